// AVOD2_79989470920838
// MI455X (gfx1250) — compile-verified
//
#include <hip/hip_runtime.h>
#include <hip/hip_bf16.h>
#include <math.h>

typedef float v2f __attribute__((ext_vector_type(2)));
typedef float v8f __attribute__((ext_vector_type(8)));

#define H_IMG 360
#define W_IMG 1200
#define H_BEV 700
#define W_BEV 800
#define C_IN  32
#define N_ANCH 32768
#define HID 256
#define TOPK 256
#define EPSBN 1e-5f
#define NEGV  -1e30f

// ---------------------------------------------------------------------------
// Kernel A: per-pixel channel dot (1x1 conv) + BN + ReLU + mask.  float4 stream.
// ---------------------------------------------------------------------------
__global__ void convbn_kernel(const float* __restrict__ fmap,
                              const float* __restrict__ w,
                              const float* __restrict__ b,
                              const float* __restrict__ bn,
                              const float* __restrict__ mask,
                              float* __restrict__ out, int HW) {
  int n4 = HW >> 2;
  int p4 = blockIdx.x * blockDim.x + threadIdx.x;
  if (p4 >= n4) return;
  const float4* f4 = (const float4*)fmap;
  float a0 = 0.f, a1 = 0.f, a2 = 0.f, a3 = 0.f;
#pragma unroll
  for (int c = 0; c < C_IN; ++c) {
    float4 v = f4[(size_t)c * n4 + p4];
    float wc = w[c];
    a0 += wc * v.x; a1 += wc * v.y; a2 += wc * v.z; a3 += wc * v.w;
  }
  float scale = rsqrtf(bn[3] + EPSBN) * bn[0];
  float shift = (b[0] - bn[2]) * scale + bn[1];
  float m = mask[0];
  float4 o;
  o.x = fmaxf(a0 * scale + shift, 0.f) * m;
  o.y = fmaxf(a1 * scale + shift, 0.f) * m;
  o.z = fmaxf(a2 * scale + shift, 0.f) * m;
  o.w = fmaxf(a3 * scale + shift, 0.f) * m;
  ((float4*)out)[p4] = o;
}

// ---------------------------------------------------------------------------
// Kernel B: ROI-align (3x3 cells, 2x2 subsamples) on both maps + fuse.
// Writes fused[n][12], K padded with zeros for WMMA (K=9 -> 12).
// ---------------------------------------------------------------------------
__device__ __forceinline__ float bilin(const float* __restrict__ f, int H, int W,
                                       float y, float x) {
  bool valid = (y > -1.0f) && (y < (float)H) && (x > -1.0f) && (x < (float)W);
  y = fminf(fmaxf(y, 0.f), (float)(H - 1));
  x = fminf(fmaxf(x, 0.f), (float)(W - 1));
  float y0 = floorf(y), x0 = floorf(x);
  int iy0 = (int)y0, ix0 = (int)x0;
  int iy1 = iy0 + 1; if (iy1 > H - 1) iy1 = H - 1;
  int ix1 = ix0 + 1; if (ix1 > W - 1) ix1 = W - 1;
  float ly = y - y0, lx = x - x0;
  float v00 = f[iy0 * W + ix0], v01 = f[iy0 * W + ix1];
  float v10 = f[iy1 * W + ix0], v11 = f[iy1 * W + ix1];
  float val = v00 * (1.f - ly) * (1.f - lx) + v01 * (1.f - ly) * lx +
              v10 * ly * (1.f - lx) + v11 * ly * lx;
  return valid ? val : 0.f;
}

__device__ __forceinline__ void roi9(const float* __restrict__ f, int H, int W,
                                     const float* roi, float* cell) {
  float x1 = roi[1] - 0.5f, y1 = roi[2] - 0.5f;
  float rw = (roi[3] - roi[1]) * (1.f / 3.f);
  float rh = (roi[4] - roi[2]) * (1.f / 3.f);
#pragma unroll
  for (int cy = 0; cy < 3; ++cy)
#pragma unroll
    for (int cx = 0; cx < 3; ++cx) {
      float acc = 0.f;
#pragma unroll
      for (int sy = 0; sy < 2; ++sy)
#pragma unroll
        for (int sx = 0; sx < 2; ++sx) {
          float tj = ((float)(cy * 2 + sy) + 0.5f) * 0.5f;
          float ti = ((float)(cx * 2 + sx) + 0.5f) * 0.5f;
          acc += bilin(f, H, W, y1 + tj * rh, x1 + ti * rw);
        }
      cell[cy * 3 + cx] = acc * 0.25f;
    }
}

__global__ void roifuse_kernel(const float* __restrict__ f_img,
                               const float* __restrict__ f_bev,
                               const float* __restrict__ anchors_img,
                               const float* __restrict__ anchors_bev,
                               const float* __restrict__ m_img,
                               const float* __restrict__ m_bev,
                               float* __restrict__ fused12) {
  int n = blockIdx.x * blockDim.x + threadIdx.x;
  if (n >= N_ANCH) return;
  float ri[5], rb[5];
#pragma unroll
  for (int j = 0; j < 5; ++j) { ri[j] = anchors_img[n * 5 + j]; rb[j] = anchors_bev[n * 5 + j]; }
  float ci[9], cb[9];
  roi9(f_img, H_IMG, W_IMG, ri, ci);
  roi9(f_bev, H_BEV, W_BEV, rb, cb);
  float inv = 1.f / (m_img[0] + m_bev[0]);
#pragma unroll
  for (int j = 0; j < 9; ++j) fused12[n * 12 + j] = (ci[j] + cb[j]) * inv;
  fused12[n * 12 + 9] = 0.f;
  fused12[n * 12 + 10] = 0.f;
  fused12[n * 12 + 11] = 0.f;
}

// ---------------------------------------------------------------------------
// Kernel C: fused MLP with V_WMMA_F32_16X16X4_F32.
// One wave (32 threads) per 16-anchor M-tile.
//   GEMM1: fused(16x12) @ W1(12x256) -> h(16x256) in LDS  (relu + bias)
//   GEMM2: h(16x256) @ Wcat(256x16; cols 0-1 obj, 2-7 off, 8-15 zero)
//   Epilogue: box decode, bev boxes, softmax score.
// ---------------------------------------------------------------------------
__global__ void mlp_wmma_kernel(const float* __restrict__ fused12,
                                const float* __restrict__ W1,
                                const float* __restrict__ b1,
                                const float* __restrict__ Wobj,
                                const float* __restrict__ bobj,
                                const float* __restrict__ Woff,
                                const float* __restrict__ boff,
                                const float* __restrict__ filtered,
                                float* __restrict__ regobj,
                                float* __restrict__ boxes,
                                float* __restrict__ scores) {
  __shared__ float hs[16 * 256];   // h tile, row-major [m][k]
  __shared__ float wc[256 * 16];   // Wcat padded [k][n], n 8..15 zero
  __shared__ float ot[16 * 8];     // output tile [m][n]

  int lane = threadIdx.x;          // 0..31 (wave32)
  int base_m = blockIdx.x * 16;
  int mrow = lane & 15;
  int khalf = (lane >> 4) << 1;    // 0 for lanes 0-15, 2 for lanes 16-31

  // Stage Wcat (zero-padded) into LDS: 4096 entries / 32 lanes = 128 iters.
  for (int idx = lane; idx < 256 * 16; idx += 32) {
    int k = idx >> 4, n = idx & 15;
    float v = 0.f;
    if (n < 2)      v = Wobj[k * 2 + n];
    else if (n < 8) v = Woff[k * 6 + (n - 2)];
    wc[idx] = v;
  }
  __syncthreads();

  // ---- GEMM1: K padded to 12; fused rows 9..11 are zero so B rows >=9 are
  // multiplied by zero (clamp B index to stay in-bounds, no divergence). ----
  for (int nt = 0; nt < 16; ++nt) {
    v8f c = {0.f, 0.f, 0.f, 0.f, 0.f, 0.f, 0.f, 0.f};
#pragma unroll
    for (int kk = 0; kk < 12; kk += 4) {
      int k0 = kk + khalf;
      v2f a, b;
      a.x = fused12[(base_m + mrow) * 12 + k0];
      a.y = fused12[(base_m + mrow) * 12 + k0 + 1];
      int n = nt * 16 + mrow;
      int kb0 = k0 < 8 ? k0 : 8;          // clamp (A is zero there anyway)
      int kb1 = (k0 + 1) < 8 ? (k0 + 1) : 8;
      b.x = W1[kb0 * 256 + n];
      b.y = W1[kb1 * 256 + n];
      c = __builtin_amdgcn_wmma_f32_16x16x4_f32(false, a, false, b,
                                                (short)0, c, false, false);
    }
    float bias = b1[nt * 16 + mrow];
#pragma unroll
    for (int r = 0; r < 8; ++r) {
      int row = (lane < 16) ? r : (r + 8);
      float v = c[r] + bias;
      hs[row * 256 + nt * 16 + mrow] = v > 0.f ? v : 0.f;
    }
  }
  __syncthreads();

  // ---- GEMM2: 16x16 out = h(16x256) @ wc(256x16), 64 K-steps ----
  v8f c2 = {0.f, 0.f, 0.f, 0.f, 0.f, 0.f, 0.f, 0.f};
  for (int kk = 0; kk < 256; kk += 4) {
    int k0 = kk + khalf;
    v2f a, b;
    a.x = hs[mrow * 256 + k0];
    a.y = hs[mrow * 256 + k0 + 1];
    b.x = wc[k0 * 16 + mrow];
    b.y = wc[(k0 + 1) * 16 + mrow];
    c2 = __builtin_amdgcn_wmma_f32_16x16x4_f32(false, a, false, b,
                                               (short)0, c2, false, false);
  }
#pragma unroll
  for (int r = 0; r < 8; ++r) {
    int row = (lane < 16) ? r : (r + 8);
    if (mrow < 8) {
      float bias = (mrow < 2) ? bobj[mrow] : boff[mrow - 2];
      ot[row * 8 + mrow] = c2[r] + bias;
    }
  }
  __syncthreads();

  // ---- Epilogue: decode 16 anchors (lanes 0..15) ----
  if (lane < 16) {
    int n = base_m + lane;
    float o0 = ot[lane * 8 + 0];
    float o1 = ot[lane * 8 + 1];
    float f0 = ot[lane * 8 + 2], f1 = ot[lane * 8 + 3], f2 = ot[lane * 8 + 4];
    float f3 = ot[lane * 8 + 5], f4 = ot[lane * 8 + 6], f5 = ot[lane * 8 + 7];
    const float* fa = filtered + (size_t)n * 6;
    float xa = fa[0], ya = fa[1], za = fa[2];
    float dxa = fa[3], dya = fa[4], dza = fa[5];
    float r0 = f0 * dxa + xa;
    float r1 = f1 * dya + ya;
    float r2 = f2 * dza + za;
    float r3 = expf(f3) * dxa;
    float r4 = expf(f4) * dya;
    float r5 = expf(f5) * dza;
    regobj[n * 8 + 0] = r0; regobj[n * 8 + 1] = r1; regobj[n * 8 + 2] = r2;
    regobj[n * 8 + 3] = r3; regobj[n * 8 + 4] = r4; regobj[n * 8 + 5] = r5;
    regobj[n * 8 + 6] = o0; regobj[n * 8 + 7] = o1;
    float4 bx;
    bx.x = r0 - r3 * 0.5f + 40.f;   // x1
    bx.y = r2 - r5 * 0.5f;          // y1
    bx.z = r0 + r3 * 0.5f + 40.f;   // x2
    bx.w = r2 + r5 * 0.5f;          // y2
    ((float4*)boxes)[n] = bx;
    float mx = fmaxf(o0, o1);
    float e0 = expf(o0 - mx), e1 = expf(o1 - mx);
    scores[n] = e1 / (e0 + e1);
  }
}

// ---------------------------------------------------------------------------
// Kernel D: sequential NMS (256 rounds), one 1024-thread block.
// Scores register-resident (32/thread, strided ownership j = t + k*1024);
// argmax via LDS tree reduce with lowest-index tie-break (matches jnp.argmax).
// Winner is self-suppressed by its own IoU (~1 > 0.8). Output gathered inline.
// ---------------------------------------------------------------------------
__global__ void __launch_bounds__(1024)
nms_kernel(const float* __restrict__ scores_g,
           const float* __restrict__ boxes,
           const float* __restrict__ regobj,
           float* __restrict__ out) {
  __shared__ float rv[1024];
  __shared__ int   ri[1024];
  int t = threadIdx.x;

  float s[32];
#pragma unroll
  for (int k = 0; k < 32; ++k) s[k] = scores_g[t + (k << 10)];

  const float4* b4 = (const float4*)boxes;

  for (int it = 0; it < TOPK; ++it) {
    // local argmax (ascending index -> strictly-greater keeps lowest idx)
    float best = -INFINITY; int bi = 0;
#pragma unroll
    for (int k = 0; k < 32; ++k) {
      if (s[k] > best) { best = s[k]; bi = t + (k << 10); }
    }
    rv[t] = best; ri[t] = bi;
    __syncthreads();
    for (int stp = 512; stp > 0; stp >>= 1) {
      if (t < stp) {
        float ov = rv[t + stp]; int oi = ri[t + stp];
        if (ov > rv[t] || (ov == rv[t] && oi < ri[t])) { rv[t] = ov; ri[t] = oi; }
      }
      __syncthreads();
    }
    int idx = ri[0];

    if (t < 8) out[it * 8 + t] = regobj[idx * 8 + t];

    float4 W = b4[idx];
    float A = (W.z - W.x) * (W.w - W.y);
#pragma unroll
    for (int k = 0; k < 32; ++k) {
      float4 bj = b4[t + (k << 10)];
      float iw = fmaxf(fminf(bj.z, W.z) - fmaxf(bj.x, W.x), 0.f);
      float ih = fmaxf(fminf(bj.w, W.w) - fmaxf(bj.y, W.y), 0.f);
      float inter = iw * ih;
      float aj = (bj.z - bj.x) * (bj.w - bj.y);
      float iou = inter / (aj + A - inter + 1e-8f);
      if (iou > 0.8f) s[k] = NEGV;
    }
    __syncthreads();   // ri[0] consumed before next round overwrites rv/ri
  }
}

// ---------------------------------------------------------------------------
extern "C" void kernel_launch(void* const* d_in, const int* in_sizes, int n_in,
                              void* d_out, int out_size, void* d_ws, size_t ws_size,
                              hipStream_t stream) {
  const float* img_map     = (const float*)d_in[0];
  const float* bev_map     = (const float*)d_in[1];
  const float* anchors_img = (const float*)d_in[2];
  const float* anchors_bev = (const float*)d_in[3];
  const float* filtered    = (const float*)d_in[4];
  const float* img_mask    = (const float*)d_in[5];
  const float* bev_mask    = (const float*)d_in[6];
  const float* w_img       = (const float*)d_in[7];
  const float* b_img       = (const float*)d_in[8];
  const float* bn_img      = (const float*)d_in[9];
  const float* w_bev       = (const float*)d_in[10];
  const float* b_bev       = (const float*)d_in[11];
  const float* bn_bev      = (const float*)d_in[12];
  const float* W1          = (const float*)d_in[13];
  const float* b1          = (const float*)d_in[14];
  const float* Wobj        = (const float*)d_in[15];
  const float* bobj        = (const float*)d_in[16];
  const float* Woff        = (const float*)d_in[17];
  const float* boff        = (const float*)d_in[18];

  float* ws     = (float*)d_ws;
  float* f_img  = ws;                       // 432000
  float* f_bev  = f_img + H_IMG * W_IMG;    // 560000
  float* fused  = f_bev + H_BEV * W_BEV;    // 32768*12
  float* regobj = fused + N_ANCH * 12;      // 32768*8
  float* boxes  = regobj + N_ANCH * 8;      // 32768*4 (16B aligned)
  float* scores = boxes + N_ANCH * 4;       // 32768

  int n4i = (H_IMG * W_IMG) / 4;
  int n4b = (H_BEV * W_BEV) / 4;
  convbn_kernel<<<(n4i + 255) / 256, 256, 0, stream>>>(
      img_map, w_img, b_img, bn_img, img_mask, f_img, H_IMG * W_IMG);
  convbn_kernel<<<(n4b + 255) / 256, 256, 0, stream>>>(
      bev_map, w_bev, b_bev, bn_bev, bev_mask, f_bev, H_BEV * W_BEV);

  roifuse_kernel<<<N_ANCH / 256, 256, 0, stream>>>(
      f_img, f_bev, anchors_img, anchors_bev, img_mask, bev_mask, fused);

  mlp_wmma_kernel<<<N_ANCH / 16, 32, 0, stream>>>(
      fused, W1, b1, Wobj, bobj, Woff, boff, filtered, regobj, boxes, scores);

  nms_kernel<<<1, 1024, 0, stream>>>(scores, boxes, regobj, (float*)d_out);
}